// Model_16475494548225
// MI455X (gfx1250) — compile-verified
//
#include <hip/hip_runtime.h>
#include <hip/hip_bf16.h>
#include <math.h>

typedef __attribute__((ext_vector_type(2))) float v2f;
typedef __attribute__((ext_vector_type(8))) float v8f;
typedef unsigned int u32x4 __attribute__((ext_vector_type(4)));
typedef int i32x4 __attribute__((ext_vector_type(4)));
typedef int i32x8 __attribute__((ext_vector_type(8)));

#define N_NODES 50000
#define N_EDGES 800000
#define D_IN 512

#if defined(__has_builtin)
#if __has_builtin(__builtin_amdgcn_tensor_load_to_lds) && \
    __has_builtin(__builtin_amdgcn_s_wait_tensorcnt)
#define USE_TDM 1
#endif
#endif
#ifndef USE_TDM
#define USE_TDM 0
#endif

// Enforce the TDM path on the DEVICE pass only. The host pass of the HIP
// dual compilation has no amdgcn builtins and legitimately type-checks the
// fallback branch (its kernel-body codegen is discarded). Round-4 probe
// proved: host pass -> USE_TDM==0, device (gfx1250) pass -> USE_TDM==1.
#if defined(__HIP_DEVICE_COMPILE__) && !USE_TDM
#error "CDNA5: tensor_load_to_lds builtin missing in DEVICE compilation"
#endif

// ---------------------------------------------------------------------------
// 1) Per-node GEMVs: app_s = x @ W_app[:512], app_d = x @ W_app[512:],
//    geo_s/geo_d from coords. One wave32 per node.
// ---------------------------------------------------------------------------
__global__ void node_feats_kernel(const float* __restrict__ x,
                                  const float* __restrict__ coords,
                                  const float* __restrict__ W_app,
                                  const float* __restrict__ W_geom,
                                  float* __restrict__ app_s, float* __restrict__ app_d,
                                  float* __restrict__ geo_s, float* __restrict__ geo_d,
                                  int n) {
  int node = (int)((blockIdx.x * blockDim.x + threadIdx.x) >> 5);
  int lane = threadIdx.x & 31;
  if (node >= n) return;
  const float* xr = x + (size_t)node * D_IN;
  float ss = 0.f, sd = 0.f;
#pragma unroll
  for (int j = 0; j < D_IN / 32; ++j) {
    int c = lane + 32 * j;
    float v = xr[c];
    ss += v * W_app[c];
    sd += v * W_app[D_IN + c];
  }
#pragma unroll
  for (int m = 16; m >= 1; m >>= 1) {
    ss += __shfl_xor(ss, m);
    sd += __shfl_xor(sd, m);
  }
  if (lane == 0) {
    app_s[node] = ss;
    app_d[node] = sd;
    float gs = 0.f, gd = 0.f;
#pragma unroll
    for (int j = 0; j < 4; ++j) {
      float cv = coords[(size_t)node * 4 + j];
      gs += cv * W_geom[j];
      gd += cv * W_geom[4 + j];
    }
    geo_s[node] = gs;
    geo_d[node] = gd;
  }
}

// deg arrays start at 1.0 (self loop weight 1 folded in)
__global__ void init_deg_kernel(float* __restrict__ deg1, float* __restrict__ deg2, int n) {
  int i = blockIdx.x * blockDim.x + threadIdx.x;
  if (i < n) { deg1[i] = 1.0f; deg2[i] = 1.0f; }
}

__global__ void dinv_kernel(const float* __restrict__ deg, float* __restrict__ dinv, int n) {
  int i = blockIdx.x * blockDim.x + threadIdx.x;
  if (i < n) {
    float d = deg[i];
    dinv[i] = (d > 0.f) ? rsqrtf(fmaxf(d, 1e-12f)) : 0.f;
  }
}

// ---------------------------------------------------------------------------
// 2) Per-edge affinity MLP -> edge weight; accumulate weighted in-degree.
// ---------------------------------------------------------------------------
__global__ void edge_emb_kernel(const int* __restrict__ ei,
                                const float* __restrict__ app_s, const float* __restrict__ app_d,
                                const float* __restrict__ geo_s, const float* __restrict__ geo_d,
                                const float* __restrict__ W_aff,
                                const float* __restrict__ b_app,
                                const float* __restrict__ b_geom,
                                const float* __restrict__ b_aff,
                                float* __restrict__ ew, float* __restrict__ deg1, int e_cnt) {
  int e = blockIdx.x * blockDim.x + threadIdx.x;
  if (e >= e_cnt) return;
  int s = ei[e], d = ei[e_cnt + e];
  float x1 = fmaxf(app_s[s] + app_d[d] + b_app[0], 0.f);
  float x2 = fmaxf(geo_s[s] + geo_d[d] + b_geom[0], 0.f);
  float w = fmaxf(x1 * W_aff[0] + x2 * W_aff[1] + b_aff[0], 0.f);
  ew[e] = w;
  atomicAdd(&deg1[d], w);
}

// ---------------------------------------------------------------------------
// 3) Dense GEMM C[M,N] = A[M,K] @ B[K,N] via V_WMMA_F32_16X16X4_F32.
//    Block = 128 threads = 4 waves; block covers 16 rows x 128 cols (each
//    wave register-blocks two 16x16 tiles so the A fragment feeds 2 WMMAs).
//    A tile (16 x K f32) is DMA'd into LDS by the Tensor Data Mover with
//    hardware padding of 4 dwords every 256 dwords (bank spread); compute
//    side addresses it as  row*(K + 4*K/256) + k + 4*(k>>8).
//    TDM is issued by wave 0 only (EXEC-ignored, TENSORcnt-tracked) and
//    completion is published to the block via s_wait_tensorcnt + barrier.
// ---------------------------------------------------------------------------
__global__ void gemm_wmma_f32(const float* __restrict__ A, const float* __restrict__ B,
                              float* __restrict__ C, int K, int N, int M) {
  extern __shared__ float As[];
  const int stride = K + ((K >> 8) << 2);  // K + 4 dwords per 256
  const int m0 = blockIdx.x * 16;
  const int t = threadIdx.x;

  // prefetch the B panel this block consumes (overlaps the TDM wait)
  for (int kp = t; kp < K; kp += 128)
    __builtin_prefetch(&B[(size_t)kp * N + blockIdx.y * 128], 0, 3);

#if USE_TDM
  if (t < 32) {  // wave 0 issues the tensor DMA
    unsigned long long ga = (unsigned long long)(const void*)(A + (size_t)m0 * K);
    unsigned uk = (unsigned)K, um = (unsigned)M;
    u32x4 g0;
    g0[0] = 1u;                                    // count=1, user descriptor
    g0[1] = 0u;                                    // lds_addr = dynamic-LDS base
    g0[2] = (unsigned)(ga & 0xFFFFFFFFull);        // global_addr[31:0]
    g0[3] = (unsigned)((ga >> 32) & 0x01FFFFFFull) | (2u << 30);  // addr hi | type=2
    i32x8 g1;
    g1[0] = (int)((2u << 16) |                     // data_size = 4B
                  (1u << 20) |                     // pad_enable
                  (7u << 22) |                     // pad_interval: every 256 dwords
                  (3u << 25));                     // pad_amount: 4 dwords
    g1[1] = (int)((uk & 0xFFFFu) << 16);           // tensor_dim0[15:0]
    g1[2] = (int)((uk >> 16) | ((um & 0xFFFFu) << 16));  // dim0 hi | dim1 lo
    g1[3] = (int)((um >> 16) | ((uk & 0xFFFFu) << 16));  // dim1 hi | tile_dim0=K
    g1[4] = 16;                                    // tile_dim1 = 16 rows (tile_dim2=0)
    g1[5] = (int)uk;                               // tensor_dim0_stride = K
    g1[6] = 0;
    g1[7] = 0;
    i32x4 gz = {0, 0, 0, 0};
#if defined(__clang_major__) && (__clang_major__ >= 23)
    i32x8 gz8 = {0, 0, 0, 0, 0, 0, 0, 0};
    __builtin_amdgcn_tensor_load_to_lds(g0, g1, gz, gz, gz8, 0);
#else
    __builtin_amdgcn_tensor_load_to_lds(g0, g1, gz, gz, 0);
#endif
    __builtin_amdgcn_s_wait_tensorcnt(0);
  }
#else
  {  // host-pass type-check only (device pass always takes the TDM branch)
    const int nq = K >> 2;
    for (int idx = t; idx < 16 * nq; idx += 128) {
      int row = idx / nq;
      int c4 = idx - row * nq;
      int col = c4 * 4;
      const float4 v =
          reinterpret_cast<const float4*>(A + (size_t)(m0 + row) * K)[c4];
      int kk = col + ((col >> 8) << 2);
      *reinterpret_cast<float4*>(&As[row * stride + kk]) = v;
    }
  }
#endif
  __syncthreads();

  const int wave = t >> 5;
  const int lane = t & 31;
  const int half = lane >> 4;
  const int r = lane & 15;
  const int n0 = blockIdx.y * 128 + wave * 32;  // two 16-col tiles per wave

  v8f acc0 = {};
  v8f acc1 = {};
  for (int k = 0; k < K; k += 4) {
    int koff = k + 2 * half;
    int kk = koff + ((koff >> 8) << 2);
    v2f a = *reinterpret_cast<const v2f*>(&As[r * stride + kk]);
    const float* Brow0 = B + (size_t)koff * N;
    const float* Brow1 = B + (size_t)(koff + 1) * N;
    v2f b0, b1;
    b0.x = Brow0[n0 + r];
    b0.y = Brow1[n0 + r];
    b1.x = Brow0[n0 + 16 + r];
    b1.y = Brow1[n0 + 16 + r];
    acc0 = __builtin_amdgcn_wmma_f32_16x16x4_f32(false, a, false, b0, (short)0,
                                                 acc0, false, false);
    acc1 = __builtin_amdgcn_wmma_f32_16x16x4_f32(false, a, false, b1, (short)0,
                                                 acc1, false, false);
  }
#pragma unroll
  for (int i = 0; i < 8; ++i) {
    int row = m0 + i + 8 * half;
    C[(size_t)row * N + n0 + r] = acc0[i];
    C[(size_t)row * N + n0 + 16 + r] = acc1[i];
  }
}

// ---------------------------------------------------------------------------
// 4) Aggregation: out[n] = dinv[n]^2 * h[n] (+bias)  then edge scatter-add.
// ---------------------------------------------------------------------------
__global__ void agg_init_kernel(const float* __restrict__ h, const float* __restrict__ dinv,
                                const float* __restrict__ bias, float* __restrict__ out,
                                int total, int C, int use_bias) {
  int i = blockIdx.x * blockDim.x + threadIdx.x;
  if (i >= total) return;
  int node = i / C;
  int c = i - node * C;
  float di = dinv[node];
  float v = di * di * h[i];
  if (use_bias) v += bias[c];
  out[i] = v;
}

__global__ void agg_edges_kernel(const int* __restrict__ ei, const float* __restrict__ h,
                                 const float* __restrict__ ewv, const float* __restrict__ dinv,
                                 float* __restrict__ out, int e_cnt, int C) {
  int e = (int)((blockIdx.x * blockDim.x + threadIdx.x) >> 5);
  int lane = threadIdx.x & 31;
  if (e >= e_cnt) return;
  int s = ei[e], d = ei[e_cnt + e];
  float norm = dinv[s] * ewv[e] * dinv[d];
  const float* hs = h + (size_t)s * C;
  float* od = out + (size_t)d * C;
  for (int c = lane; c < C; c += 32) atomicAdd(&od[c], norm * hs[c]);
}

// ---------------------------------------------------------------------------
// 5) bias + ReLU on out1 (in place) fused with the W_m1 GEMV halves.
// ---------------------------------------------------------------------------
__global__ void post1_kernel(float* __restrict__ out1, const float* __restrict__ b_c1,
                             const float* __restrict__ W_m1,
                             float* __restrict__ m_s, float* __restrict__ m_d, int n) {
  int node = (int)((blockIdx.x * blockDim.x + threadIdx.x) >> 5);
  int lane = threadIdx.x & 31;
  if (node >= n) return;
  float* row = out1 + (size_t)node * 256;
  float ms = 0.f, md = 0.f;
#pragma unroll
  for (int j = 0; j < 8; ++j) {
    int c = lane + 32 * j;
    float v = fmaxf(row[c] + b_c1[c], 0.f);
    row[c] = v;
    ms += v * W_m1[c];
    md += v * W_m1[256 + c];
  }
#pragma unroll
  for (int m = 16; m >= 1; m >>= 1) {
    ms += __shfl_xor(ms, m);
    md += __shfl_xor(md, m);
  }
  if (lane == 0) { m_s[node] = ms; m_d[node] = md; }
}

__global__ void edge_attr2_kernel(const int* __restrict__ ei,
                                  const float* __restrict__ m_s, const float* __restrict__ m_d,
                                  const float* __restrict__ b_m1,
                                  float* __restrict__ ea2, float* __restrict__ deg2, int e_cnt) {
  int e = blockIdx.x * blockDim.x + threadIdx.x;
  if (e >= e_cnt) return;
  int s = ei[e], d = ei[e_cnt + e];
  float v = fmaxf(m_s[s] + m_d[d] + b_m1[0], 0.f);
  ea2[e] = v;
  atomicAdd(&deg2[d], v);
}

// ---------------------------------------------------------------------------
// 6) Final pairwise classifier: 16 edges per block. diff = out2[a]-out2[b]
//    staged in LDS (gathers hit the 192MB L2: out2 is only 25.6MB),
//    16x128 @ 128x64 via WMMA, then ReLU * W_f2 row-reduce (xor-butterfly
//    stays inside each 16-lane half = one D-matrix row group), cross-wave
//    combine in LDS, sigmoid.
// ---------------------------------------------------------------------------
__global__ void classifier_wmma_kernel(const float* __restrict__ out2,
                                       const int* __restrict__ ei2,
                                       const float* __restrict__ W_f1,
                                       const float* __restrict__ b_f1,
                                       const float* __restrict__ W_f2,
                                       const float* __restrict__ b_f2,
                                       float* __restrict__ out, int e_cnt) {
  __shared__ float diff[16 * 132];   // 128 + 4 pad
  __shared__ float partial[64];      // 4 waves x 16 rows
  const int t = threadIdx.x;
  const int e0 = blockIdx.x * 16;

  for (int idx = t; idx < 16 * 32; idx += 128) {
    int row = idx >> 5;
    int c4 = idx & 31;
    int e = e0 + row;
    int a = ei2[e], b = ei2[e_cnt + e];
    float4 va = reinterpret_cast<const float4*>(out2 + (size_t)a * 128)[c4];
    float4 vb = reinterpret_cast<const float4*>(out2 + (size_t)b * 128)[c4];
    float4 v;
    v.x = va.x - vb.x; v.y = va.y - vb.y; v.z = va.z - vb.z; v.w = va.w - vb.w;
    *reinterpret_cast<float4*>(&diff[row * 132 + c4 * 4]) = v;
  }
  __syncthreads();

  const int wave = t >> 5;
  const int lane = t & 31;
  const int half = lane >> 4;
  const int r = lane & 15;
  const int n0 = wave * 16;  // hidden-unit columns for this wave

  v8f acc = {};
  for (int k = 0; k < 128; k += 4) {
    int koff = k + 2 * half;
    v2f a = *reinterpret_cast<const v2f*>(&diff[r * 132 + koff]);
    v2f b;
    b.x = W_f1[koff * 64 + n0 + r];
    b.y = W_f1[(koff + 1) * 64 + n0 + r];
    acc = __builtin_amdgcn_wmma_f32_16x16x4_f32(false, a, false, b, (short)0,
                                                acc, false, false);
  }

  const float bf = b_f1[n0 + r];
  const float wf = W_f2[n0 + r];
#pragma unroll
  for (int i = 0; i < 8; ++i) {
    float h = fmaxf(acc[i] + bf, 0.f);
    float c = h * wf;
    c += __shfl_xor(c, 1);
    c += __shfl_xor(c, 2);
    c += __shfl_xor(c, 4);
    c += __shfl_xor(c, 8);
    if (r == 0) partial[wave * 16 + 8 * half + i] = c;
  }
  __syncthreads();

  if (t < 16) {
    float logit = b_f2[0];
#pragma unroll
    for (int w = 0; w < 4; ++w) logit += partial[w * 16 + t];
    out[e0 + t] = 1.0f / (1.0f + expf(-logit));
  }
}

// ---------------------------------------------------------------------------
extern "C" void kernel_launch(void* const* d_in, const int* in_sizes, int n_in,
                              void* d_out, int out_size, void* d_ws, size_t ws_size,
                              hipStream_t stream) {
  const float* x      = (const float*)d_in[0];
  const float* coords = (const float*)d_in[1];
  const float* W_app  = (const float*)d_in[2];
  const float* b_app  = (const float*)d_in[3];
  const float* W_geom = (const float*)d_in[4];
  const float* b_geom = (const float*)d_in[5];
  const float* W_aff  = (const float*)d_in[6];
  const float* b_aff  = (const float*)d_in[7];
  const float* W_c1   = (const float*)d_in[8];
  const float* b_c1   = (const float*)d_in[9];
  const float* W_m1   = (const float*)d_in[10];
  const float* b_m1   = (const float*)d_in[11];
  const float* W_c2   = (const float*)d_in[12];
  const float* b_c2   = (const float*)d_in[13];
  const float* W_f1   = (const float*)d_in[14];
  const float* b_f1   = (const float*)d_in[15];
  const float* W_f2   = (const float*)d_in[16];
  const float* b_f2   = (const float*)d_in[17];
  const int*   ei     = (const int*)d_in[18];
  const int*   ei2    = (const int*)d_in[19];
  float* outp = (float*)d_out;

  const int N = N_NODES;
  const int E = N_EDGES;
  const int E2 = in_sizes[19] / 2;

  // workspace layout (floats). h2 aliases h1, out2 aliases out1 (sequential).
  float* ws = (float*)d_ws;
  size_t off = 0;
  float* app_s = ws + off; off += N;
  float* app_d = ws + off; off += N;
  float* geo_s = ws + off; off += N;
  float* geo_d = ws + off; off += N;
  float* deg1  = ws + off; off += N;
  float* dinv1 = ws + off; off += N;
  float* m_s   = ws + off; off += N;
  float* m_d   = ws + off; off += N;
  float* deg2  = ws + off; off += N;
  float* dinv2 = ws + off; off += N;
  float* ew    = ws + off; off += E;
  float* ea2   = ws + off; off += E;
  float* hbuf  = ws + off; off += (size_t)N * 256;  // h1, later h2 (N*128)
  float* obuf  = ws + off; off += (size_t)N * 256;  // out1, later out2 (N*128)

  float* h1 = hbuf;
  float* out1 = obuf;
  float* h2 = hbuf;   // reuse
  float* out2 = obuf; // reuse

  // 1) per-node GEMVs + degree init
  node_feats_kernel<<<(N * 32 + 255) / 256, 256, 0, stream>>>(
      x, coords, W_app, W_geom, app_s, app_d, geo_s, geo_d, N);
  init_deg_kernel<<<(N + 255) / 256, 256, 0, stream>>>(deg1, deg2, N);

  // 2) edge affinity + deg1
  edge_emb_kernel<<<(E + 255) / 256, 256, 0, stream>>>(
      ei, app_s, app_d, geo_s, geo_d, W_aff, b_app, b_geom, b_aff, ew, deg1, E);
  dinv_kernel<<<(N + 255) / 256, 256, 0, stream>>>(deg1, dinv1, N);

  // 3) GEMM1: h1 = x @ W_c1   [50000,512]x[512,256]
  {
    dim3 grid(N / 16, 256 / 128);
    size_t shmem = 16 * (512 + 8) * sizeof(float);
    gemm_wmma_f32<<<grid, 128, shmem, stream>>>(x, W_c1, h1, 512, 256, N);
  }

  // 4) aggregate 1
  agg_init_kernel<<<(N * 256 + 255) / 256, 256, 0, stream>>>(
      h1, dinv1, b_c1, out1, N * 256, 256, 0);
  agg_edges_kernel<<<(E * 32 + 255) / 256, 256, 0, stream>>>(
      ei, h1, ew, dinv1, out1, E, 256);

  // 5) bias+relu + W_m1 GEMV
  post1_kernel<<<(N * 32 + 255) / 256, 256, 0, stream>>>(
      out1, b_c1, W_m1, m_s, m_d, N);

  // 6) edge_attr2 + deg2
  edge_attr2_kernel<<<(E + 255) / 256, 256, 0, stream>>>(
      ei, m_s, m_d, b_m1, ea2, deg2, E);
  dinv_kernel<<<(N + 255) / 256, 256, 0, stream>>>(deg2, dinv2, N);

  // 7) GEMM2: h2 = out1 @ W_c2   [50000,256]x[256,128]
  {
    dim3 grid(N / 16, 128 / 128);
    size_t shmem = 16 * (256 + 4) * sizeof(float);
    gemm_wmma_f32<<<grid, 128, shmem, stream>>>(out1, W_c2, h2, 256, 128, N);
  }

  // 8) aggregate 2 (bias folded into self-loop init; no relu)
  agg_init_kernel<<<(N * 128 + 255) / 256, 256, 0, stream>>>(
      h2, dinv2, b_c2, out2, N * 128, 128, 1);
  agg_edges_kernel<<<(E * 32 + 255) / 256, 256, 0, stream>>>(
      ei, h2, ea2, dinv2, out2, E, 128);

  // 9) final pairwise classifier (WMMA) -> sigmoid
  classifier_wmma_kernel<<<E2 / 16, 128, 0, stream>>>(
      out2, ei2, W_f1, b_f1, W_f2, b_f2, outp, E2);
}